// GuardNet_28587302322281
// MI455X (gfx1250) — compile-verified
//
#include <hip/hip_runtime.h>
#include <hip/hip_bf16.h>
#include <math.h>

#define N_NODES 50000
#define N_EDGES 800000
#define D_FEAT  64
#define D_HIDE  64
#define N_CLASS 40

typedef __attribute__((ext_vector_type(2))) float v2f;
typedef __attribute__((ext_vector_type(8))) float v8f;

// ---------------- wave32 reductions ----------------
__device__ __forceinline__ float wave_sum(float v) {
#pragma unroll
    for (int off = 16; off; off >>= 1) v += __shfl_xor(v, off, 32);
    return v;
}
__device__ __forceinline__ float wave_max(float v) {
#pragma unroll
    for (int off = 16; off; off >>= 1) v = fmaxf(v, __shfl_xor(v, off, 32));
    return v;
}

// ---------------- utility ----------------
__global__ void zero_kernel(float* p, int n) {
    int i = blockIdx.x * blockDim.x + threadIdx.x;
    if (i < n) p[i] = 0.0f;
}

// zero-pad W [rows x cols] into Wp [rows x colsp]
__global__ void pad_w_kernel(const float* __restrict__ W, float* __restrict__ Wp,
                             int rows, int cols, int colsp) {
    int idx = blockIdx.x * blockDim.x + threadIdx.x;
    if (idx >= rows * colsp) return;
    int r = idx / colsp, c = idx - r * colsp;
    Wp[idx] = (c < cols) ? W[r * cols + c] : 0.0f;
}

// L2-normalize each 64-dim node row; one wave per node
__global__ void normalize_kernel(const float* __restrict__ feat,
                                 float* __restrict__ fn, int n) {
    int wid  = (blockIdx.x * blockDim.x + threadIdx.x) >> 5;
    int lane = threadIdx.x & 31;
    if (wid >= n) return;
    const float* r = feat + (size_t)wid * D_FEAT;
    float a0 = r[lane], a1 = r[lane + 32];
    float ss  = wave_sum(a0 * a0 + a1 * a1);
    float inv = 1.0f / fmaxf(sqrtf(ss), 1e-12f);
    float* o = fn + (size_t)wid * D_FEAT;
    o[lane]      = a0 * inv;
    o[lane + 32] = a1 * inv;
}

// cosine sim per edge (wave per edge), threshold, atomic row-sum + degree count
__global__ void edge_sim_kernel(const float* __restrict__ fn,
                                const long long* __restrict__ ei,
                                float* __restrict__ sim,
                                float* __restrict__ rs,
                                float* __restrict__ degc, int e_cnt) {
    int wid  = (blockIdx.x * blockDim.x + threadIdx.x) >> 5;
    int lane = threadIdx.x & 31;
    if (wid >= e_cnt) return;
    int row = (int)ei[wid];
    int col = (int)ei[(size_t)e_cnt + wid];
    const float* a = fn + (size_t)row * D_FEAT;
    const float* b = fn + (size_t)col * D_FEAT;
    float s = wave_sum(a[lane] * b[lane] + a[lane + 32] * b[lane + 32]);
    s = (s < 0.1f) ? 0.0f : s;
    if (lane == 0) {
        sim[wid] = s;
        if (s > 0.0f) {
            atomicAdd(rs + row, s);
            atomicAdd(degc + row, 1.0f);
        }
    }
}

// w_self = exp(1/(deg+1));  dgcn initialized to w_self
__global__ void self_weight_kernel(const float* __restrict__ degc,
                                   float* __restrict__ wself,
                                   float* __restrict__ dgcn, int n) {
    int i = blockIdx.x * blockDim.x + threadIdx.x;
    if (i >= n) return;
    float ws = __expf(1.0f / (degc[i] + 1.0f));
    wself[i] = ws;
    dgcn[i]  = ws;
}

// w_edge = exp(sim / max(rs[row],eps)) for sim>0; accumulate weighted degree
__global__ void edge_weight_kernel(const long long* __restrict__ ei,
                                   float* __restrict__ sim,
                                   const float* __restrict__ rs,
                                   float* __restrict__ dgcn, int e_cnt) {
    int e = blockIdx.x * blockDim.x + threadIdx.x;
    if (e >= e_cnt) return;
    float s = sim[e];
    if (s > 0.0f) {
        int row = (int)ei[e];
        float w = __expf(s / fmaxf(rs[row], 1e-12f));
        sim[e] = w;
        atomicAdd(dgcn + row, w);
    } else {
        sim[e] = 0.0f;
    }
}

__global__ void dinv_kernel(const float* __restrict__ dgcn,
                            float* __restrict__ dinv, int n) {
    int i = blockIdx.x * blockDim.x + threadIdx.x;
    if (i >= n) return;
    dinv[i] = rsqrtf(fmaxf(dgcn[i], 1e-12f));
}

// ---------------- WMMA f32 GEMM: C[M x NACT] = A[M x 64] * Bp[64 x NCPAD] ----
// NCPAD/NACT are compile-time: unconditional loads, immediate-offset address
// math, no exec-mask churn in the inner loop. One wave per 16x16 output tile,
// 16 chained v_wmma_f32_16x16x4_f32.
template <int NCPAD, int NACT>
__global__ void gemm_wmma_kernel(const float* __restrict__ A,
                                 const float* __restrict__ Bp,
                                 float* __restrict__ C, int m_rows) {
    constexpr int TN = NCPAD / 16;
    int wid  = (blockIdx.x * blockDim.x + threadIdx.x) >> 5;
    int lane = threadIdx.x & 31;
    int tiles_m = m_rows >> 4;
    if (wid >= tiles_m * TN) return;    // whole-wave exit: EXEC all-1 at WMMA
    int tm = wid / TN;
    int tn = wid - tm * TN;

    int half = lane >> 4;               // 0: K pair {k,k+1}; 1: {k+2,k+3}
    int l16  = lane & 15;
    int mrow = tm * 16 + l16;           // A row held by this lane
    int ncol = tn * 16 + l16;           // B col held by this lane

    const float* Aptr = A  + (size_t)mrow * D_FEAT + half * 2;
    const float* Bptr = Bp + (size_t)(half * 2) * NCPAD + ncol;

    v8f acc = {};
#pragma unroll
    for (int k = 0; k < D_FEAT; k += 4) {
        v2f a, b;
        a.x = Aptr[k];
        a.y = Aptr[k + 1];
        b.x = Bptr[k * NCPAD];
        b.y = Bptr[(k + 1) * NCPAD];
        acc = __builtin_amdgcn_wmma_f32_16x16x4_f32(
            /*neg_a=*/false, a, /*neg_b=*/false, b,
            /*c_mod=*/(short)0, acc, /*reuse_a=*/false, /*reuse_b=*/false);
    }
    // D layout: VGPR i -> row (tm*16 + half*8 + i), col ncol
    if (NACT == NCPAD || ncol < NACT) {
#pragma unroll
        for (int i = 0; i < 8; ++i)
            C[(size_t)(tm * 16 + half * 8 + i) * NACT + ncol] = acc[i];
    }
}

// scatter: acc[row,f] += w_edge * dinv[row]*dinv[col] * hw[col,f]
__global__ void edge_scatter_kernel(const long long* __restrict__ ei,
                                    const float* __restrict__ w_edge,
                                    const float* __restrict__ dinv,
                                    const float* __restrict__ hw,
                                    float* __restrict__ acc,
                                    int e_cnt, int fout) {
    int idx = blockIdx.x * blockDim.x + threadIdx.x;
    int tot = e_cnt * fout;
    if (idx >= tot) return;
    int e = idx / fout;
    int f = idx - e * fout;
    float w = w_edge[e];
    if (w == 0.0f) return;
    int row = (int)ei[e];
    int col = (int)ei[(size_t)e_cnt + e];
    float coef = w * dinv[row] * dinv[col];
    atomicAdd(acc + (size_t)row * fout + f, coef * hw[(size_t)col * fout + f]);
}

// out = acc + w_self*dinv^2 * hw + b  (optional relu)
__global__ void finalize_kernel(const float* __restrict__ accb,
                                const float* __restrict__ hw,
                                const float* __restrict__ wself,
                                const float* __restrict__ dinv,
                                const float* __restrict__ bias,
                                float* __restrict__ out,
                                int n, int fout, int do_relu) {
    int idx = blockIdx.x * blockDim.x + threadIdx.x;
    int tot = n * fout;
    if (idx >= tot) return;
    int i = idx / fout;
    int f = idx - i * fout;
    float di = dinv[i];
    float v = accb[idx] + wself[i] * di * di * hw[idx] + bias[f];
    out[idx] = do_relu ? fmaxf(v, 0.0f) : v;
}

// in-place log_softmax over N_CLASS columns; one wave per node
__global__ void log_softmax_kernel(float* __restrict__ out, int n) {
    int wid  = (blockIdx.x * blockDim.x + threadIdx.x) >> 5;
    int lane = threadIdx.x & 31;
    if (wid >= n) return;
    float* r = out + (size_t)wid * N_CLASS;
    const float NEG = -3.0e38f;
    float v0 = (lane < N_CLASS) ? r[lane] : NEG;
    float v1 = (lane + 32 < N_CLASS) ? r[lane + 32] : NEG;
    float m = wave_max(fmaxf(v0, v1));
    float s = wave_sum(((lane < N_CLASS) ? __expf(v0 - m) : 0.0f) +
                       ((lane + 32 < N_CLASS) ? __expf(v1 - m) : 0.0f));
    float l = __logf(s);
    if (lane < N_CLASS)       r[lane]      = v0 - m - l;
    if (lane + 32 < N_CLASS)  r[lane + 32] = v1 - m - l;
}

// ---------------- host-side layer driver ----------------
static void run_layer(const float* feat, const long long* ei,
                      const float* Bpad, const float* bias,
                      float* out, int fout, int do_relu,
                      float* fn, float* hw, float* accb, float* sim,
                      float* rs, float* degc, float* wself, float* dgcn, float* dinv,
                      hipStream_t stream) {
    const int T = 256;
    const int n = N_NODES, e = N_EDGES;

    zero_kernel<<<(2 * n + T - 1) / T, T, 0, stream>>>(rs, 2 * n);   // rs + degc contiguous
    normalize_kernel<<<(n * 32 + T - 1) / T, T, 0, stream>>>(feat, fn, n);
    edge_sim_kernel<<<(e * 32 + T - 1) / T, T, 0, stream>>>(fn, ei, sim, rs, degc, e);
    self_weight_kernel<<<(n + T - 1) / T, T, 0, stream>>>(degc, wself, dgcn, n);
    edge_weight_kernel<<<(e + T - 1) / T, T, 0, stream>>>(ei, sim, rs, dgcn, e);
    dinv_kernel<<<(n + T - 1) / T, T, 0, stream>>>(dgcn, dinv, n);

    if (fout == D_HIDE) {
        int tiles = (n / 16) * (64 / 16);
        gemm_wmma_kernel<64, 64><<<(tiles * 32 + T - 1) / T, T, 0, stream>>>(feat, Bpad, hw, n);
    } else {
        int tiles = (n / 16) * (48 / 16);
        gemm_wmma_kernel<48, 40><<<(tiles * 32 + T - 1) / T, T, 0, stream>>>(feat, Bpad, hw, n);
    }

    zero_kernel<<<(n * fout + T - 1) / T, T, 0, stream>>>(accb, n * fout);
    edge_scatter_kernel<<<(e * fout + T - 1) / T, T, 0, stream>>>(ei, sim, dinv, hw, accb, e, fout);
    finalize_kernel<<<(n * fout + T - 1) / T, T, 0, stream>>>(accb, hw, wself, dinv, bias, out,
                                                              n, fout, do_relu);
}

extern "C" void kernel_launch(void* const* d_in, const int* in_sizes, int n_in,
                              void* d_out, int out_size, void* d_ws, size_t ws_size,
                              hipStream_t stream) {
    const float*     x  = (const float*)d_in[0];
    const long long* ei = (const long long*)d_in[1];   // int64 edge_index [2, E]
    const float*     W1 = (const float*)d_in[2];
    const float*     b1 = (const float*)d_in[3];
    const float*     W2 = (const float*)d_in[4];
    const float*     b2 = (const float*)d_in[5];
    float*           out = (float*)d_out;

    // workspace layout (floats)
    float* ws    = (float*)d_ws;
    size_t NF    = (size_t)N_NODES * D_FEAT;
    float* fn    = ws;              // N*64  normalized features
    float* h     = fn + NF;         // N*64  hidden layer output
    float* hw    = h + NF;          // N*64  GEMM product (layer2 uses N*40)
    float* accb  = hw + NF;         // N*64  segment-sum accumulator
    float* sim   = accb + NF;       // E     sim -> w_edge (in place)
    float* rs    = sim + N_EDGES;   // N     row sums   } contiguous for one zero pass
    float* degc  = rs + N_NODES;    // N     degree cnt }
    float* wself = degc + N_NODES;  // N
    float* dgcn  = wself + N_NODES; // N
    float* dinv  = dgcn + N_NODES;  // N
    float* w2p   = dinv + N_NODES;  // 64*48 zero-padded W2

    const int T = 256;
    pad_w_kernel<<<(64 * 48 + T - 1) / T, T, 0, stream>>>(W2, w2p, 64, N_CLASS, 48);

    // layer 1: attention on raw x, GEMM x@W1 (already 64x64, no pad), relu -> h
    run_layer(x, ei, W1, b1, h, D_HIDE, /*relu=*/1,
              fn, hw, accb, sim, rs, degc, wself, dgcn, dinv, stream);
    // layer 2: attention on h, GEMM h@W2pad -> d_out
    run_layer(h, ei, w2p, b2, out, N_CLASS, /*relu=*/0,
              fn, hw, accb, sim, rs, degc, wself, dgcn, dinv, stream);

    log_softmax_kernel<<<(N_NODES * 32 + T - 1) / T, T, 0, stream>>>(out, N_NODES);
}